// CLCNetStep_23708219474742
// MI455X (gfx1250) — compile-verified
//
#include <hip/hip_runtime.h>

// ---------------------------------------------------------------------------
// CLCNet single step for gfx1250 (MI455X).
// Bandwidth-bound matvec chain (~102 MB fp32 weights, ~51 MFLOP => 0.5 FLOP/B).
// Strategy: V_WMMA_F32_16X16X4_F32 matvec tiles (exact fp32 math), one b64
// weight load per lane per WMMA, x vector staged in LDS, split-K=4 partials
// with deterministic reduction (no atomics). Weights stay L2-resident (192MB).
// ---------------------------------------------------------------------------

typedef float v2f __attribute__((ext_vector_type(2)));
typedef float v8f __attribute__((ext_vector_type(8)));

#define O_    5
#define F_    161
#define H_    2048
#define IN_   322          // 2*F
#define OUTN_ 3220         // O*F*4
#define EPS_    1e-8f
#define ALPHA_  0.99f
#define BNEPS_  1e-5f

// workspace float offsets
#define V0_   0            // 336 (322 + zero pad)
#define V1_   336          // 336 (322 + zero pad)
#define GI_   672          // 6144
#define GHP_  6816         // 4 * 6144 split partials of W_hh @ h
#define HNEW_ 31392        // 2048
#define Y2P_  33440        // 4 * 2048 split partials of W2 @ h_new
#define U_    41632        // 2048
#define Y3P_  43680        // 4 * 3220 split partials of Wout @ u
// total = 56560 floats = 226,240 bytes of d_ws

__device__ __forceinline__ v8f zero8() {
  v8f z;
#pragma unroll
  for (int i = 0; i < 8; ++i) z[i] = 0.0f;
  return z;
}

__device__ __forceinline__ v8f wmma_step(v2f a, v2f b, v8f c) {
  // V_WMMA_F32_16X16X4_F32: D = A(16x4) * B(4x16) + C; we use column 0 of B/D.
  return __builtin_amdgcn_wmma_f32_16x16x4_f32(false, a, false, b, (short)0, c,
                                               false, false);
}

// Accumulate nk4*4 contraction elements starting at global k index kbeg.
// xl points at an LDS copy of x[kbeg .. kbeg + nk4*4).  K must be %4 == 0 here.
__device__ __forceinline__ v8f mv_acc(const float* __restrict__ W, int ld,
                                      const float* xl, int kbeg, int nk4,
                                      int row, v8f c) {
  const int lane = threadIdx.x & 31;
  const int koff = (lane >> 4) * 2;                 // 0 (lanes 0-15) / 2 (16-31)
  const float* wp = W + (size_t)row * (size_t)ld + kbeg + koff;
  const float* xp = xl + koff;
#pragma unroll 4
  for (int j = 0; j < nk4; ++j) {
    v2f a = *(const v2f*)wp;                        // global b64, weight stream
    v2f b = *(const v2f*)xp;                        // LDS b64
    c = wmma_step(a, b, c);
    wp += 4;
    xp += 4;
  }
  return c;
}

// Same but with a K tail (K even, K % 4 == 2): zero the out-of-range k pair.
__device__ __forceinline__ v8f mv_acc_tail(const float* __restrict__ W, int ld,
                                           int K, const float* xl, int row,
                                           v8f c) {
  const int lane = threadIdx.x & 31;
  const int koff = (lane >> 4) * 2;
  const int nk4 = (K + 3) >> 2;
  const float* wrow = W + (size_t)row * (size_t)ld;
#pragma unroll 4
  for (int j = 0; j < nk4; ++j) {
    int kk = j * 4 + koff;
    v2f a = {0.0f, 0.0f};
    v2f b = {0.0f, 0.0f};
    if (kk + 1 < K) {                               // kk,kk+1 both valid
      a = *(const v2f*)(wrow + kk);
      b = *(const v2f*)(xl + kk);
    }
    c = wmma_step(a, b, c);
  }
  return c;
}

// --- kernel 1: magnitude scan, h_norm_new, normalized input vector v0 -------
__global__ void k_prep(const float* __restrict__ x, const float* __restrict__ hn,
                       float* __restrict__ out, float* __restrict__ ws) {
  int tid = threadIdx.x;
  if (tid < 14) {                                   // zero pads (322..335)
    ws[V0_ + 322 + tid] = 0.0f;
    ws[V1_ + 322 + tid] = 0.0f;
  }
  if (tid < F_) {
    float s = hn[tid];
#pragma unroll
    for (int t = 0; t < O_; ++t) {
      float xr = x[t * (2 * F_) + 2 * tid];
      float xi = x[t * (2 * F_) + 2 * tid + 1];
      float xm = sqrtf(xr * xr + xi * xi + EPS_);
      s = (1.0f - ALPHA_) * xm + ALPHA_ * s;
    }
    out[322 + tid] = s;                             // h_norm_new
    float inv = 1.0f / ((s + EPS_) * 50.0f);
    float xr = x[4 * (2 * F_) + 2 * tid];
    float xi = x[4 * (2 * F_) + 2 * tid + 1];
    ws[V0_ + 2 * tid] = xr * inv;
    ws[V0_ + 2 * tid + 1] = xi * inv;
  }
}

// --- kernel 2: fc1 (322x322) + BN + relu -> v1 ------------------------------
__global__ void k_fc1(const float* __restrict__ W1, const float* __restrict__ b1,
                      const float* __restrict__ g1, const float* __restrict__ be1,
                      const float* __restrict__ rm1, const float* __restrict__ rv1,
                      float* __restrict__ ws) {
  __shared__ float xl[324];
  int tid = threadIdx.x;
  for (int i = tid; i < 324; i += 256) xl[i] = ws[V0_ + i];
  __syncthreads();
  int lane = tid & 31;
  int rowbase = (blockIdx.x * 8 + (tid >> 5)) * 16;
  int row = rowbase + (lane & 15);
  if (row > IN_ - 1) row = IN_ - 1;
  v8f c = zero8();
  c = mv_acc_tail(W1, IN_, IN_, xl, row, c);
  if ((lane & 15) == 0) {
    int rb = rowbase + (lane >> 4) * 8;
#pragma unroll
    for (int i = 0; i < 8; ++i) {
      int r = rb + i;
      if (r < IN_) {
        float y = c[i] + b1[r];
        y = (y - rm1[r]) * (g1[r] / sqrtf(rv1[r] + BNEPS_)) + be1[r];
        ws[V1_ + r] = fmaxf(y, 0.0f);
      }
    }
  }
}

// --- kernel 3: GRU matvecs: gi = Wih@v1, ghp[s] = (Whh@h) split partials ----
__global__ void k_gru_mv(const float* __restrict__ Wih,
                         const float* __restrict__ Whh,
                         const float* __restrict__ h, float* __restrict__ ws) {
  __shared__ float xi[324];
  __shared__ float xh[512];
  int tid = threadIdx.x;
  int s = blockIdx.y;                               // K split 0..3
  if (s == 0)
    for (int i = tid; i < 324; i += 256) xi[i] = ws[V1_ + i];
  for (int i = tid; i < 512; i += 256) xh[i] = h[s * 512 + i];
  __syncthreads();
  int lane = tid & 31;
  int rowbase = (blockIdx.x * 8 + (tid >> 5)) * 16; // < 6144
  int row = rowbase + (lane & 15);
  if (s == 0) {                                     // block-uniform branch
    v8f c = zero8();
    c = mv_acc_tail(Wih, IN_, IN_, xi, row, c);
    if ((lane & 15) == 0) {
      int rb = rowbase + (lane >> 4) * 8;
#pragma unroll
      for (int i = 0; i < 8; ++i) ws[GI_ + rb + i] = c[i];
    }
  }
  v8f c2 = zero8();
  c2 = mv_acc(Whh, H_, xh, s * 512, 128, row, c2);
  if ((lane & 15) == 0) {
    int rb = rowbase + (lane >> 4) * 8;
#pragma unroll
    for (int i = 0; i < 8; ++i) ws[GHP_ + s * 6144 + rb + i] = c2[i];
  }
}

// --- kernel 4: GRU gates -> h_new -------------------------------------------
__global__ void k_gate(const float* __restrict__ bih, const float* __restrict__ bhh,
                       const float* __restrict__ h, float* __restrict__ out,
                       float* __restrict__ ws) {
  int i = blockIdx.x * blockDim.x + threadIdx.x;
  if (i >= H_) return;
  float gir = ws[GI_ + i] + bih[i];
  float giz = ws[GI_ + H_ + i] + bih[H_ + i];
  float gin = ws[GI_ + 2 * H_ + i] + bih[2 * H_ + i];
  float ghr = bhh[i], ghz = bhh[H_ + i], ghn = bhh[2 * H_ + i];
#pragma unroll
  for (int s = 0; s < 4; ++s) {                     // deterministic split sum
    const float* gp = ws + GHP_ + s * 6144;
    ghr += gp[i];
    ghz += gp[H_ + i];
    ghn += gp[2 * H_ + i];
  }
  float r = 1.0f / (1.0f + expf(-(gir + ghr)));
  float z = 1.0f / (1.0f + expf(-(giz + ghz)));
  float n = tanhf(gin + r * ghn);
  float hnv = (1.0f - z) * n + z * h[i];
  ws[HNEW_ + i] = hnv;
  out[483 + i] = hnv;                               // h_new output
}

// --- kernel 5: fc2 matvec split partials ------------------------------------
__global__ void k_fc2_mv(const float* __restrict__ W2, float* __restrict__ ws) {
  __shared__ float xh[512];
  int tid = threadIdx.x;
  int s = blockIdx.y;
  for (int i = tid; i < 512; i += 256) xh[i] = ws[HNEW_ + s * 512 + i];
  __syncthreads();
  int lane = tid & 31;
  int rowbase = (blockIdx.x * 8 + (tid >> 5)) * 16; // < 2048
  int row = rowbase + (lane & 15);
  v8f c = zero8();
  c = mv_acc(W2, H_, xh, s * 512, 128, row, c);
  if ((lane & 15) == 0) {
    int rb = rowbase + (lane >> 4) * 8;
#pragma unroll
    for (int i = 0; i < 8; ++i) ws[Y2P_ + s * H_ + rb + i] = c[i];
  }
}

// --- kernel 6: fc2 reduce + BN + relu -> u ----------------------------------
__global__ void k_fc2_ep(const float* __restrict__ b2, const float* __restrict__ g2,
                         const float* __restrict__ be2, const float* __restrict__ rm2,
                         const float* __restrict__ rv2, float* __restrict__ ws) {
  int i = blockIdx.x * blockDim.x + threadIdx.x;
  if (i >= H_) return;
  float y = b2[i];
#pragma unroll
  for (int s = 0; s < 4; ++s) y += ws[Y2P_ + s * H_ + i];
  y = (y - rm2[i]) * (g2[i] / sqrtf(rv2[i] + BNEPS_)) + be2[i];
  ws[U_ + i] = fmaxf(y, 0.0f);
}

// --- kernel 7: fc_out matvec split partials ---------------------------------
__global__ void k_out_mv(const float* __restrict__ Wo, float* __restrict__ ws) {
  __shared__ float xu[512];
  int tid = threadIdx.x;
  int s = blockIdx.y;
  for (int i = tid; i < 512; i += 256) xu[i] = ws[U_ + s * 512 + i];
  __syncthreads();
  int lane = tid & 31;
  int rowbase = (blockIdx.x * 8 + (tid >> 5)) * 16;
  int row = rowbase + (lane & 15);
  if (row > OUTN_ - 1) row = OUTN_ - 1;
  v8f c = zero8();
  c = mv_acc(Wo, H_, xu, s * 512, 128, row, c);
  if ((lane & 15) == 0) {
    int rb = rowbase + (lane >> 4) * 8;
#pragma unroll
    for (int i = 0; i < 8; ++i) {
      int r = rb + i;
      if (r < OUTN_) ws[Y3P_ + s * OUTN_ + r] = c[i];
    }
  }
}

// --- kernel 8: reduce + tanh*2 coef, complex combine -> out[F,2] ------------
__global__ void k_final(const float* __restrict__ x, const float* __restrict__ bo,
                        float* __restrict__ out, const float* __restrict__ ws) {
  int f = threadIdx.x;
  if (f >= F_) return;
  float re = 0.0f, im = 0.0f;
#pragma unroll
  for (int o = 0; o < O_; ++o) {
    int idx = o * (2 * F_) + 2 * f;
    float yr = bo[idx], yi = bo[idx + 1];
#pragma unroll
    for (int s = 0; s < 4; ++s) {
      yr += ws[Y3P_ + s * OUTN_ + idx];
      yi += ws[Y3P_ + s * OUTN_ + idx + 1];
    }
    float cr = tanhf(yr) * 2.0f;
    float ci = tanhf(yi) * 2.0f;
    float xr = x[idx], xi2 = x[idx + 1];
    re += xr * cr - xi2 * ci;
    im += xr * ci + xi2 * cr;
  }
  out[2 * f] = re;
  out[2 * f + 1] = im;
}

extern "C" void kernel_launch(void* const* d_in, const int* in_sizes, int n_in,
                              void* d_out, int out_size, void* d_ws, size_t ws_size,
                              hipStream_t stream) {
  (void)in_sizes; (void)n_in; (void)out_size; (void)ws_size;
  const float* x   = (const float*)d_in[0];
  const float* hn  = (const float*)d_in[1];
  const float* hr  = (const float*)d_in[2];
  const float* W1  = (const float*)d_in[3];
  const float* b1  = (const float*)d_in[4];
  const float* g1  = (const float*)d_in[5];
  const float* be1 = (const float*)d_in[6];
  const float* rm1 = (const float*)d_in[7];
  const float* rv1 = (const float*)d_in[8];
  const float* Wih = (const float*)d_in[9];
  const float* Whh = (const float*)d_in[10];
  const float* bih = (const float*)d_in[11];
  const float* bhh = (const float*)d_in[12];
  const float* W2  = (const float*)d_in[13];
  const float* b2  = (const float*)d_in[14];
  const float* g2  = (const float*)d_in[15];
  const float* be2 = (const float*)d_in[16];
  const float* rm2 = (const float*)d_in[17];
  const float* rv2 = (const float*)d_in[18];
  const float* Wo  = (const float*)d_in[19];
  const float* bo  = (const float*)d_in[20];
  float* out = (float*)d_out;
  float* ws  = (float*)d_ws;

  k_prep<<<1, 256, 0, stream>>>(x, hn, out, ws);
  k_fc1<<<3, 256, 0, stream>>>(W1, b1, g1, be1, rm1, rv1, ws);
  k_gru_mv<<<dim3(48, 4), 256, 0, stream>>>(Wih, Whh, hr, ws);
  k_gate<<<8, 256, 0, stream>>>(bih, bhh, hr, out, ws);
  k_fc2_mv<<<dim3(16, 4), 256, 0, stream>>>(W2, ws);
  k_fc2_ep<<<8, 256, 0, stream>>>(b2, g2, be2, rm2, rv2, ws);
  k_out_mv<<<dim3(26, 4), 256, 0, stream>>>(Wo, ws);
  k_final<<<1, 192, 0, stream>>>(x, bo, out, ws);
}